// SODENModel_54743653155347
// MI455X (gfx1250) — compile-verified
//
#include <hip/hip_runtime.h>

typedef __attribute__((ext_vector_type(2))) float v2f;
typedef __attribute__((ext_vector_type(8))) float v8f;

#define NEVAL 33   // eval points u = j*(h/2)*t, j=0..32

// Single v_max_num_f32 with inline-constant 0; bypasses the compiler's
// canonicalize (v_max x,x) it otherwise inserts for opaque WMMA results.
__device__ __forceinline__ float relu_asm(float x) {
    float r;
    asm("v_max_num_f32 %0, 0, %1" : "=v"(r) : "v"(x));
    return r;
}

__global__ __launch_bounds__(256) void soden_kernel(
    const float* __restrict__ t, const float* __restrict__ init_cond,
    const float* __restrict__ features, const float* __restrict__ W1,
    const float* __restrict__ b1, const float* __restrict__ W2,
    const float* __restrict__ b2, const float* __restrict__ W3,
    const float* __restrict__ b3, const float* __restrict__ beta,
    float* __restrict__ out, int n)
{
    __shared__ float sW1[32], sb1[32], sb2[32], sW3[32], sb3s[1];
    __shared__ __align__(16) float sW2[1024];
    __shared__ __align__(16) float sBeta[128];

    const int tid = threadIdx.x;
    if (tid < 32) { sW1[tid] = W1[tid]; sb1[tid] = b1[tid];
                    sb2[tid] = b2[tid]; sW3[tid] = W3[tid]; }
    for (int i = tid; i < 1024; i += 256) sW2[i] = W2[i];
    if (tid < 128) sBeta[tid] = beta[tid];
    if (tid == 0)  sb3s[0] = b3[0];
    __syncthreads();

    const int lane = tid & 31;
    const int wid  = tid >> 5;
    const int half = lane >> 4;   // which K-pair half of the A/B slice
    const int lo   = lane & 15;   // row (sample) / column index
    int sample = blockIdx.x * 128 + wid * 16 + lo;
    if (sample >= n) sample = n - 1;          // clamp keeps EXEC all-ones for WMMA

    // ---- per-lane weight registers (ISA 16x16x4 f32 A/B layouts) ----
    float w1a[8], w1b[8], b1a[8], b1b[8];
    v2f bw0[8], bw1[8];
#pragma unroll
    for (int kk = 0; kk < 8; ++kk) {
        const int k0 = 4 * kk + 2 * half;
        w1a[kk] = sW1[k0];     w1b[kk] = sW1[k0 + 1];
        b1a[kk] = sb1[k0];     b1b[kk] = sb1[k0 + 1];
        bw0[kk].x = sW2[k0 * 32 + lo];        // B tile, cols 0..15
        bw0[kk].y = sW2[(k0 + 1) * 32 + lo];
        bw1[kk].x = sW2[k0 * 32 + 16 + lo];   // B tile, cols 16..31
        bw1[kk].y = sW2[(k0 + 1) * 32 + 16 + lo];
    }
    const float b2t0 = sb2[lo], b2t1 = sb2[16 + lo];
    const float w3a  = sW3[lo], w3b  = sW3[16 + lo];
    const float b3v  = sb3s[0];

    const float t_s = t[sample];
    float accw = 0.f;     // sum of w_j * softplus_j  (scaled at the end)
    float lamv = 0.f;
    const float hstep = 1.0f / 16.0f;

    for (int j = 0; j < NEVAL; ++j) {
        const float x = (0.5f * hstep * (float)j) * t_s;

        v8f c0, c1;                         // C/D: bias preloaded
#pragma unroll
        for (int r = 0; r < 8; ++r) { c0[r] = b2t0; c1[r] = b2t1; }

#pragma unroll
        for (int kk = 0; kk < 8; ++kk) {    // K = 32 in 8 slices of 4
            v2f a;
            a.x = fmaxf(fmaf(x, w1a[kk], b1a[kk]), 0.f);  // h1 entries
            a.y = fmaxf(fmaf(x, w1b[kk], b1b[kk]), 0.f);
            c0 = __builtin_amdgcn_wmma_f32_16x16x4_f32(
                     false, a, false, bw0[kk], (short)0, c0, false, false);
            c1 = __builtin_amdgcn_wmma_f32_16x16x4_f32(
                     false, a, false, bw1[kk], (short)0, c1, false, false);
        }

        // relu(h2) (1 instr each), then per-lane partial y = h2 @ W3
        float p[8];
#pragma unroll
        for (int r = 0; r < 8; ++r) {
            const float d0 = relu_asm(c0[r]);
            const float d1 = relu_asm(c1[r]);
            p[r] = fmaf(d0, w3a, d1 * w3b);
        }

        // Register-merging butterfly: 16-lane sums of all 8 partials in
        // 8 shuffles + 8 adds. After stage 3, each lane holds the 8-lane
        // sum of p[lane&7]; final xor-8 completes the 16-lane sum.
        const int b0 = lane & 1, b1_ = lane & 2, b2_ = lane & 4;
        float k0 = b0 ? p[1] : p[0];  float g0 = b0 ? p[0] : p[1];
        float k1 = b0 ? p[3] : p[2];  float g1 = b0 ? p[2] : p[3];
        float k2 = b0 ? p[5] : p[4];  float g2 = b0 ? p[4] : p[5];
        float k3 = b0 ? p[7] : p[6];  float g3 = b0 ? p[6] : p[7];
        float z0 = k0 + __shfl_xor(g0, 1, 32);   // pairsum of p[bit0]
        float z1 = k1 + __shfl_xor(g1, 1, 32);   // pairsum of p[2+bit0]
        float z2 = k2 + __shfl_xor(g2, 1, 32);   // pairsum of p[4+bit0]
        float z3 = k3 + __shfl_xor(g3, 1, 32);   // pairsum of p[6+bit0]
        float k4 = b1_ ? z1 : z0;  float g4 = b1_ ? z0 : z1;
        float k5 = b1_ ? z3 : z2;  float g5 = b1_ ? z2 : z3;
        float w0 = k4 + __shfl_xor(g4, 2, 32);   // 4-sum of p[lane&3]
        float w1s = k5 + __shfl_xor(g5, 2, 32);  // 4-sum of p[4+(lane&3)]
        float k6 = b2_ ? w1s : w0; float g6 = b2_ ? w0 : w1s;
        float v0 = k6 + __shfl_xor(g6, 4, 32);   // 8-sum of p[lane&7]
        float S  = v0 + __shfl_xor(v0, 8, 32);   // 16-sum: y_{(lane&7)+8*(lane>>4)}

        // route y_m to lane m (and m+16): source lane (lane&7) + 16*bit3(lane)
        const float y = __shfl(S, (lane & 7) + ((lane & 8) << 1), 32);

        const float z  = y + b3v;
        const float sp = fmaxf(z, 0.f) + log1pf(expf(-fabsf(z)));  // softplus
        const float w  = (j == 0 || j == 32) ? 1.0f : ((j & 1) ? 4.0f : 2.0f);
        accw = fmaf(w, sp, accw);                                   // Simpson sum
        if (j == NEVAL - 1) lamv = sp;                              // lam = g(t)
    }

    // Lam = init_cond + (h/6) * t * sum(w_j * sp_j)   [f = sp * t]
    const float acc = fmaf((hstep / 6.0f) * t_s, accw, init_cond[sample]);

    // ---- prod = features @ beta, split across lane halves ----
    const float4* frow = (const float4*)(features + (size_t)sample * 128);
    const float4* bet4 = (const float4*)sBeta;
    float part = 0.f;
    const int off = half * 16;      // 64 floats per half, in float4 units
#pragma unroll 4
    for (int q = 0; q < 16; ++q) {
        const float4 v = frow[off + q];
        const float4 w4 = bet4[off + q];
        part = fmaf(v.x, w4.x, part);
        part = fmaf(v.y, w4.y, part);
        part = fmaf(v.z, w4.z, part);
        part = fmaf(v.w, w4.w, part);
    }
    const float prod = part + __shfl_xor(part, 16, 32);

    const float prod_exp = expf(fminf(prod, 10.0f));
    const float Lambda = acc * prod_exp;
    const float lamo   = lamv * prod_exp;
    const float logl   = logf(fmaxf(lamv, 1e-8f)) + prod;

    if (lane < 16) {
        const int s = blockIdx.x * 128 + wid * 16 + lo;
        if (s < n) {
            out[s]         = Lambda;
            out[n + s]     = lamo;
            out[2 * n + s] = logl;
        }
    }
}

extern "C" void kernel_launch(void* const* d_in, const int* in_sizes, int n_in,
                              void* d_out, int out_size, void* d_ws, size_t ws_size,
                              hipStream_t stream) {
    const float* t    = (const float*)d_in[0];
    const float* ic   = (const float*)d_in[1];
    const float* feat = (const float*)d_in[2];
    const float* W1   = (const float*)d_in[3];
    const float* b1   = (const float*)d_in[4];
    const float* W2   = (const float*)d_in[5];
    const float* b2   = (const float*)d_in[6];
    const float* W3   = (const float*)d_in[7];
    const float* b3   = (const float*)d_in[8];
    const float* beta = (const float*)d_in[9];

    const int n = in_sizes[0];
    const int blocks = (n + 127) / 128;   // 128 samples per 256-thread block
    soden_kernel<<<blocks, 256, 0, stream>>>(t, ic, feat, W1, b1, W2, b2, W3, b3,
                                             beta, (float*)d_out, n);
}